// SA_v3_39797166964879
// MI455X (gfx1250) — compile-verified
//
#include <hip/hip_runtime.h>
#include <math.h>

typedef __attribute__((ext_vector_type(2))) float v2f;
typedef __attribute__((ext_vector_type(8))) float v8f;

#define BATCH 16
#define CH    64
#define HH    256
#define WW_   256
#define WIN   32     // window height/width (256/8)
#define NTOK  64     // 8*8 windows
#define DK    64

// Async 128-bit global -> LDS copy (CDNA5, ASYNCcnt-tracked, no VGPR round-trip).
// lds_off: LDS byte address (low 32 bits of a generic pointer to __shared__).
__device__ __forceinline__ void async_b128(unsigned lds_off, const void* gaddr) {
  asm volatile("global_load_async_to_lds_b128 %0, %1, off"
               :: "v"(lds_off), "v"(gaddr) : "memory");
}
__device__ __forceinline__ void wait_async0() {
  asm volatile("s_wait_asynccnt 0" ::: "memory");
}

// ---------------- Kernel 1: adaptive avg pool -> y[b][n][c] ----------------
// One block per (b, c, window). Exactly one float4 load per thread:
// 256 threads x 4 floats = the full 32x32 window, 128B-coalesced per row.
__global__ void pool_kernel(const float* __restrict__ x, float* __restrict__ y) {
  int idx = blockIdx.x;                 // b*CH*NTOK + c*NTOK + n
  int n = idx & 63;
  int c = (idx >> 6) & 63;
  int b = idx >> 12;
  int mi = n >> 3, mj = n & 7;
  const float* base = x + (((size_t)b * CH + c) * HH + mi * WIN) * WW_ + mj * WIN;
  int t = threadIdx.x;
  int row = t >> 3, c4 = t & 7;
  float4 v = *(const float4*)(base + (size_t)row * WW_ + c4 * 4);
  float s = v.x + v.y + v.z + v.w;
  for (int off = 16; off > 0; off >>= 1) s += __shfl_down(s, off, 32);
  __shared__ float red[8];
  int lane = t & 31, wv = t >> 5;
  if (lane == 0) red[wv] = s;
  __syncthreads();
  if (t == 0) {
    float acc = 0.f;
#pragma unroll
    for (int i = 0; i < 8; ++i) acc += red[i];
    y[((size_t)b * NTOK + n) * CH + c] = acc * (1.0f / (WIN * WIN));
  }
}

// ------------- Kernel 2: q,k projection + scores + softmax -> dist ---------
// One block per batch. Tiny (64x64 matrices); negligible runtime.
__global__ void score_kernel(const float* __restrict__ y,
                             const float* __restrict__ Wq,
                             const float* __restrict__ Wk,
                             float* __restrict__ dist) {
  __shared__ float sy[NTOK * CH];   // 16 KB
  __shared__ float sq[NTOK * DK];   // 16 KB
  __shared__ float sk[NTOK * DK];   // 16 KB
  int b = blockIdx.x;
  int t = threadIdx.x;
  for (int i = t; i < NTOK * CH; i += 256) sy[i] = y[(size_t)b * NTOK * CH + i];
  __syncthreads();
  for (int i = t; i < NTOK * DK; i += 256) {
    int nn = i >> 6, d = i & 63;
    float aq = 0.f, ak = 0.f;
    for (int cc = 0; cc < CH; ++cc) {
      float yv = sy[nn * CH + cc];
      aq += yv * Wq[d * CH + cc];
      ak += yv * Wk[d * CH + cc];
    }
    sq[i] = aq;
    sk[i] = ak;
  }
  __syncthreads();
  // scores (reuse sy)
  for (int i = t; i < NTOK * NTOK; i += 256) {
    int nn = i >> 6, m = i & 63;
    float acc = 0.f;
    for (int d = 0; d < DK; ++d) acc += sq[nn * DK + d] * sk[m * DK + d];
    sy[i] = acc * 0.125f;             // 1/sqrt(64)
  }
  __syncthreads();
  if (t < NTOK) {
    float mx = -INFINITY;
    for (int m = 0; m < NTOK; ++m) mx = fmaxf(mx, sy[t * NTOK + m]);
    float sum = 0.f;
    for (int m = 0; m < NTOK; ++m) { float e = __expf(sy[t * NTOK + m] - mx); sy[t * NTOK + m] = e; sum += e; }
    float inv = 1.0f / sum;
    for (int m = 0; m < NTOK; ++m) dist[(size_t)b * NTOK * NTOK + t * NTOK + m] = sy[t * NTOK + m] * inv;
  }
}

// ---------------- Kernel 3: out = dist @ V via V_WMMA_F32_16X16X4_F32 ------
// One block per (b, c, i-row-in-window). dist(64x64) and the V(64x32) slice are
// staged into LDS with async B128 copies; 8 waves = 4 n-tiles x 2 j-tiles, each
// wave accumulating a 16x16 f32 tile over K=64 with 16 WMMA ops.
__global__ void att_kernel(const float* __restrict__ x,
                           const float* __restrict__ dist,
                           float* __restrict__ out) {
  __shared__ __align__(16) float sD[NTOK * NTOK];   // 16 KB: dist for this batch
  __shared__ __align__(16) float sV[NTOK * WIN];    // 8 KB : V[m][j] slice for (b,c,i)
  int idx = blockIdx.x;               // b*CH*WIN + c*WIN + i
  int i = idx & 31;
  int c = (idx >> 5) & 63;
  int b = idx >> 11;
  int t = threadIdx.x;

  // Stage dist: 1024 float4 -> 4 async B128 copies per thread.
  const float4* gD = (const float4*)(dist + (size_t)b * NTOK * NTOK);
#pragma unroll
  for (int r = 0; r < 4; ++r) {
    int k4 = t + r * 256;             // 0..1023
    async_b128((unsigned)(uintptr_t)&sD[k4 * 4], gD + k4);
  }
  // Stage V: V[m][j] = x[b,c,(m/8)*32+i,(m%8)*32+j]; 512 float4 -> 2 per thread.
#pragma unroll
  for (int r = 0; r < 2; ++r) {
    int k4 = t + r * 256;             // 0..511 : m = k4>>3, j4 = k4&7
    int m = k4 >> 3, j4 = k4 & 7;
    const float4* g = (const float4*)(x + (((size_t)b * CH + c) * HH + (m >> 3) * WIN + i) * WW_
                                        + (m & 7) * WIN + j4 * 4);
    async_b128((unsigned)(uintptr_t)&sV[k4 * 4], g);
  }
  wait_async0();        // this wave's copies complete
  __syncthreads();      // all waves' copies visible

  int lane = t & 31, half = lane >> 4, lm = lane & 15;
  int wv = t >> 5;
  int nt = wv >> 1;                   // token tile 0..3
  int jt = wv & 1;                    // j tile 0..1
  v8f acc = {};
#pragma unroll
  for (int kk = 0; kk < 16; ++kk) {
    // A: 16x4 f32, vgpr j holds K = j + 2*(lane/16), row M = lane%16
    v2f a, bf;
    a[0] = sD[(nt * 16 + lm) * NTOK + kk * 4 + 0 + 2 * half];
    a[1] = sD[(nt * 16 + lm) * NTOK + kk * 4 + 1 + 2 * half];
    // B: 4x16 f32, vgpr r holds K-row r + 2*(lane/16), col N = lane%16
    bf[0] = sV[(kk * 4 + 0 + 2 * half) * WIN + jt * 16 + lm];
    bf[1] = sV[(kk * 4 + 1 + 2 * half) * WIN + jt * 16 + lm];
    acc = __builtin_amdgcn_wmma_f32_16x16x4_f32(
        /*neg_a=*/false, a, /*neg_b=*/false, bf,
        /*c_mod=*/(short)0, acc, /*reuse_a=*/false, /*reuse_b=*/false);
  }

  // C/D layout: vgpr v -> row M = v + 8*(lane/16), col N = lane%16
#pragma unroll
  for (int v = 0; v < 8; ++v) {
    int n = nt * 16 + v + 8 * half;
    int j = jt * 16 + lm;
    out[(((size_t)b * CH + c) * HH + (n >> 3) * WIN + i) * WW_ + (n & 7) * WIN + j] = acc[v];
  }
}

extern "C" void kernel_launch(void* const* d_in, const int* in_sizes, int n_in,
                              void* d_out, int out_size, void* d_ws, size_t ws_size,
                              hipStream_t stream) {
  const float* x  = (const float*)d_in[0];
  const float* Wq = (const float*)d_in[1];
  const float* Wk = (const float*)d_in[2];
  float* out  = (float*)d_out;
  float* y    = (float*)d_ws;                          // 16*64*64 floats (256 KB)
  float* dist = (float*)d_ws + BATCH * NTOK * CH;      // 16*64*64 floats (256 KB)

  pool_kernel <<<BATCH * CH * NTOK, 256, 0, stream>>>(x, y);
  score_kernel<<<BATCH,             256, 0, stream>>>(y, Wq, Wk, dist);
  att_kernel  <<<BATCH * CH * WIN,  256, 0, stream>>>(x, dist, out);
}